// GATFraudGNN_91018946937013
// MI455X (gfx1250) — compile-verified
//
#include <hip/hip_runtime.h>
#include <hip/hip_bf16.h>

typedef __attribute__((ext_vector_type(2))) float v2f;
typedef __attribute__((ext_vector_type(8))) float v8f;

#define NNODES    50000
#define NEDGES    400000
#define NEG_SLOPE 0.2f

// ---------------------------------------------------------------- utilities

__global__ __launch_bounds__(256) void fill_kernel(float* p, float v, long n) {
  long i = (long)blockIdx.x * blockDim.x + threadIdx.x;
  if (i < n) p[i] = v;
}

__device__ __forceinline__ float atomicMaxFloat(float* addr, float value) {
  // monotone int mapping trick; buffer must be initialized to -inf
  if (value >= 0.0f)
    return __int_as_float(atomicMax((int*)addr, __float_as_int(value)));
  else
    return __uint_as_float(atomicMin((unsigned int*)addr, __float_as_uint(value)));
}

// ------------------------------------------------- WMMA f32 GEMM  C = A @ B
// A: [M,K] row-major (M multiple of 16, K multiple of 4), B: [K,N] row-major,
// C: [M,N]. Block = 256 threads = 8 waves; block covers one 16-row M tile and
// up to 8 x 16 = 128 columns. The 16 x K A tile is staged into LDS with
// GLOBAL_LOAD_ASYNC_TO_LDS_B128 (ASYNCcnt-tracked DMA, no VGPR round trip).
__global__ __launch_bounds__(256) void wmma_gemm_f32(
    const float* __restrict__ A, const float* __restrict__ B,
    float* __restrict__ C, int M, int N, int K) {
  extern __shared__ float ldsA[];            // 16 * K floats
  const int m0 = blockIdx.x * 16;

  // async stage of the A tile: 16B per lane per issue
  for (int i = threadIdx.x * 4; i < 16 * K; i += blockDim.x * 4) {
    int r = i / K, c = i - r * K;            // 4-float chunk never crosses a row
    const float* gptr = &A[(long)(m0 + r) * K + c];
    unsigned lds_byte = (unsigned)i * 4u;    // dynamic LDS starts at offset 0
    asm volatile("global_load_async_to_lds_b128 %0, %1, off"
                 :: "v"(lds_byte), "v"(gptr) : "memory");
  }
  asm volatile("s_wait_asynccnt 0x0" ::: "memory");
  __syncthreads();

  const int wave = threadIdx.x >> 5;
  const int n0 = blockIdx.y * 128 + wave * 16;
  if (n0 >= N) return;

  const int lane = threadIdx.x & 31;
  const int half = lane >> 4;                // 0: lanes 0-15, 1: lanes 16-31
  const int lrow = lane & 15;

  v8f acc = {};
  for (int k = 0; k < K; k += 4) {
    v2f a, b;
    // A 16x4 frag: VGPR0 = K={k, k+2}, VGPR1 = K={k+1, k+3} across halves
    a.x = ldsA[lrow * K + k + half * 2 + 0];
    a.y = ldsA[lrow * K + k + half * 2 + 1];
    // B 4x16 frag: rows striped across lanes within a VGPR
    b.x = B[(long)(k + half * 2 + 0) * N + n0 + lrow];
    b.y = B[(long)(k + half * 2 + 1) * N + n0 + lrow];
    acc = __builtin_amdgcn_wmma_f32_16x16x4_f32(
        /*neg_a=*/false, a, /*neg_b=*/false, b,
        /*c_mod=*/(short)0, acc, /*reuse_a=*/false, /*reuse_b=*/false);
  }
  // C 16x16 frag: VGPR j holds row (half*8 + j), col = lrow
#pragma unroll
  for (int j = 0; j < 8; ++j)
    C[(long)(m0 + half * 8 + j) * N + n0 + lrow] = acc[j];
}

// ----------------------------------------- per-(node,head) attention logits
// one wave per (node, head); h row = 64 contiguous floats
__global__ __launch_bounds__(256) void att_logits_kernel(
    const float* __restrict__ h, const float* __restrict__ a_src,
    const float* __restrict__ a_dst, float* __restrict__ al_src,
    float* __restrict__ al_dst, int n_nodes, int heads) {
  long wid = ((long)blockIdx.x * blockDim.x + threadIdx.x) >> 5;
  int lane = threadIdx.x & 31;
  long total = (long)n_nodes * heads;
  if (wid >= total) return;
  int hd = (int)(wid % heads);
  const float* row = h + wid * 64;
  float s = 0.f, d = 0.f;
#pragma unroll
  for (int c = lane; c < 64; c += 32) {
    float v = row[c];
    s += v * a_src[hd * 64 + c];
    d += v * a_dst[hd * 64 + c];
  }
#pragma unroll
  for (int off = 16; off; off >>= 1) {
    s += __shfl_xor(s, off);
    d += __shfl_xor(d, off);
  }
  if (lane == 0) { al_src[wid] = s; al_dst[wid] = d; }
}

// ------------------------------------------------- pass A: segment max of e
__global__ __launch_bounds__(256) void edge_max_kernel(
    const int* __restrict__ src, const int* __restrict__ dst, int e_base,
    const float* __restrict__ al_src, const float* __restrict__ al_dst,
    float* __restrict__ emax, int heads, long total) {
  long idx = (long)blockIdx.x * blockDim.x + threadIdx.x;
  if (idx >= total) return;
  long e = idx / heads;
  int hd = (int)(idx - e * heads);
  int s, d;
  if (e < e_base) { s = src[e]; d = dst[e]; }
  else            { s = d = (int)(e - e_base); }        // self-loop
  float v = al_src[(long)s * heads + hd] + al_dst[(long)d * heads + hd];
  v = v > 0.f ? v : NEG_SLOPE * v;                      // leaky relu
  atomicMaxFloat(&emax[(long)d * heads + hd], v);
}

// --------------------- pass B: exp, denom atomic-add, weighted scatter-add
// one wave per (edge, head); 64 channels -> 2 atomics per lane
__global__ __launch_bounds__(256) void edge_agg_kernel(
    const int* __restrict__ src, const int* __restrict__ dst, int e_base,
    const float* __restrict__ h, const float* __restrict__ al_src,
    const float* __restrict__ al_dst, const float* __restrict__ emax,
    float* __restrict__ denom, float* __restrict__ agg, int heads,
    long total_waves) {
  long wid = ((long)blockIdx.x * blockDim.x + threadIdx.x) >> 5;
  int lane = threadIdx.x & 31;
  if (wid >= total_waves) return;
  long e = wid / heads;
  int hd = (int)(wid - e * heads);
  int s, d;
  if (e < e_base) { s = src[e]; d = dst[e]; }
  else            { s = d = (int)(e - e_base); }
  const float* hrow = h + ((long)s * heads + hd) * 64;   // 256B contiguous
  // overlap the random gather with the logit/exp computation
  __builtin_prefetch(&hrow[lane], 0, 0);
  __builtin_prefetch(&hrow[lane + 32], 0, 0);
  float v = al_src[(long)s * heads + hd] + al_dst[(long)d * heads + hd];
  v = v > 0.f ? v : NEG_SLOPE * v;
  float ex = __expf(v - emax[(long)d * heads + hd]);
  if (lane == 0) atomicAdd(&denom[(long)d * heads + hd], ex);
  float* arow = agg + ((long)d * heads + hd) * 64;
  atomicAdd(&arow[lane],      hrow[lane]      * ex);
  atomicAdd(&arow[lane + 32], hrow[lane + 32] * ex);
}

// ------------------------------- pass C: divide by denom, add bias, ReLU
__global__ __launch_bounds__(256) void finalize_kernel(
    float* __restrict__ agg, const float* __restrict__ denom,
    const float* __restrict__ bias, int heads, long total) {
  long i = (long)blockIdx.x * blockDim.x + threadIdx.x;
  if (i >= total) return;
  long nh = i >> 6;                 // (node, head)
  int c = (int)(i & 63);
  int hd = (int)(nh % heads);
  float v = agg[i] / denom[nh] + bias[hd * 64 + c];
  agg[i] = v > 0.f ? v : 0.f;
}

// --------------------------------------------- classifier: out = h@Wc + bc
__global__ __launch_bounds__(256) void classify_kernel(
    const float* __restrict__ h, const float* __restrict__ Wc,
    const float* __restrict__ bc, float* __restrict__ out, int n_nodes) {
  long wid = ((long)blockIdx.x * blockDim.x + threadIdx.x) >> 5;
  int lane = threadIdx.x & 31;
  if (wid >= n_nodes) return;
  const float* row = h + wid * 64;
  float s = 0.f;
#pragma unroll
  for (int c = lane; c < 64; c += 32) s += row[c] * Wc[c];
#pragma unroll
  for (int off = 16; off; off >>= 1) s += __shfl_xor(s, off);
  if (lane == 0) out[wid] = s + bc[0];
}

// ---------------------------------------------------------------- launcher

static inline unsigned cdiv_u(long a, long b) { return (unsigned)((a + b - 1) / b); }

extern "C" void kernel_launch(void* const* d_in, const int* in_sizes, int n_in,
                              void* d_out, int out_size, void* d_ws, size_t ws_size,
                              hipStream_t stream) {
  const float* x      = (const float*)d_in[0];   // [N,128]
  const int*   eidx   = (const int*)  d_in[1];   // [2,E]
  const float* W1     = (const float*)d_in[2];   // [128,512]
  const float* a_src1 = (const float*)d_in[3];   // [8,64]
  const float* a_dst1 = (const float*)d_in[4];
  const float* b1     = (const float*)d_in[5];   // [512]
  const float* W2     = (const float*)d_in[6];   // [512,64]
  const float* a_src2 = (const float*)d_in[7];   // [64]
  const float* a_dst2 = (const float*)d_in[8];
  const float* b2     = (const float*)d_in[9];   // [64]
  const float* Wc     = (const float*)d_in[10];  // [64,1]
  const float* bc     = (const float*)d_in[11];  // [1]
  float* out = (float*)d_out;

  const int  N = NNODES, E = NEDGES;
  const int* src = eidx;
  const int* dst = eidx + E;
  const long EP  = (long)E + N;                  // edges incl. self-loops

  // ---- workspace layout (floats) ----
  float* ws = (float*)d_ws;
  size_t o = 0;
  float* h1   = ws + o; o += (size_t)N * 512;    // x@W1      (recycled later)
  float* agg1 = ws + o; o += (size_t)N * 512;    // layer-1 aggregate / h after relu
  float* als1 = ws + o; o += (size_t)N * 8;
  float* ald1 = ws + o; o += (size_t)N * 8;
  float* emx1 = ws + o; o += (size_t)N * 8;
  float* den1 = ws + o; o += (size_t)N * 8;
  // layer-2 buffers recycle the h1 region after finalize1
  float* h2   = h1;                              // [N,64]
  float* agg2 = h1 + (size_t)N * 64;             // [N,64]
  float* als2 = h1 + (size_t)N * 128;
  float* ald2 = als2 + N;
  float* emx2 = ald2 + N;
  float* den2 = emx2 + N;

  const float NEG_INF = -__builtin_huge_valf();

  // ================= layer 1 (8 heads x 64 ch, concat) =================
  fill_kernel<<<cdiv_u((long)N * 512, 256), 256, 0, stream>>>(agg1, 0.f, (long)N * 512);
  fill_kernel<<<cdiv_u((long)N * 8, 256), 256, 0, stream>>>(den1, 0.f, (long)N * 8);
  fill_kernel<<<cdiv_u((long)N * 8, 256), 256, 0, stream>>>(emx1, NEG_INF, (long)N * 8);

  {
    dim3 grid(N / 16, 512 / 128);
    wmma_gemm_f32<<<grid, 256, 16 * 128 * sizeof(float), stream>>>(x, W1, h1, N, 512, 128);
  }
  att_logits_kernel<<<cdiv_u((long)N * 8, 8), 256, 0, stream>>>(h1, a_src1, a_dst1, als1, ald1, N, 8);
  edge_max_kernel<<<cdiv_u(EP * 8, 256), 256, 0, stream>>>(src, dst, E, als1, ald1, emx1, 8, EP * 8);
  edge_agg_kernel<<<cdiv_u(EP * 8, 8), 256, 0, stream>>>(src, dst, E, h1, als1, ald1, emx1, den1, agg1, 8, EP * 8);
  finalize_kernel<<<cdiv_u((long)N * 512, 256), 256, 0, stream>>>(agg1, den1, b1, 8, (long)N * 512);

  // ================= layer 2 (1 head x 64 ch, mean==identity) ==========
  fill_kernel<<<cdiv_u((long)N * 64, 256), 256, 0, stream>>>(agg2, 0.f, (long)N * 64);
  fill_kernel<<<cdiv_u((long)N, 256), 256, 0, stream>>>(den2, 0.f, (long)N);
  fill_kernel<<<cdiv_u((long)N, 256), 256, 0, stream>>>(emx2, NEG_INF, (long)N);

  {
    dim3 grid(N / 16, 1);   // 64 cols -> 4 active waves of 8
    wmma_gemm_f32<<<grid, 256, 16 * 512 * sizeof(float), stream>>>(agg1, W2, h2, N, 64, 512);
  }
  att_logits_kernel<<<cdiv_u((long)N, 8), 256, 0, stream>>>(h2, a_src2, a_dst2, als2, ald2, N, 1);
  edge_max_kernel<<<cdiv_u(EP, 256), 256, 0, stream>>>(src, dst, E, als2, ald2, emx2, 1, EP);
  edge_agg_kernel<<<cdiv_u(EP, 8), 256, 0, stream>>>(src, dst, E, h2, als2, ald2, emx2, den2, agg2, 1, EP);
  finalize_kernel<<<cdiv_u((long)N * 64, 256), 256, 0, stream>>>(agg2, den2, b2, 1, (long)N * 64);

  // ================= classifier =================
  classify_kernel<<<cdiv_u((long)N, 8), 256, 0, stream>>>(agg2, Wc, bc, out, N);
}